// LegendreNetwork1D_5342939316638
// MI455X (gfx1250) — compile-verified
//
#include <hip/hip_runtime.h>
#include <hip/hip_bf16.h>
#include <stdint.h>

#define DEGREE 20
#define NW (DEGREE + 1)
#define BLOCK 256

typedef float v4f __attribute__((ext_vector_type(4)));

// Monic Legendre recurrence: pi_n = x*pi_{n-1} - c_n*pi_{n-2},
// c_n = (n-1)^2 / ((2n-1)(2n-3)); P_n = K_n * pi_n with K_n = (2n-1)!!/n!
// folded into the weights (SALU s_mul_f32 at kernel entry), so the fused
// series costs 3 VALU ops/degree (mul + fma + accumulate-fma).
__device__ __forceinline__ float leg_eval(float x, const float* __restrict__ wp) {
    float pm1 = 1.0f;                                   // pi_0
    float p   = x;                                      // pi_1
    float acc = __builtin_fmaf(wp[1], x, wp[0]);
#pragma unroll
    for (int n = 2; n <= DEGREE; ++n) {
        const float c = (float)(((double)(n - 1) * (double)(n - 1)) /
                                ((2.0 * n - 1.0) * (2.0 * n - 3.0)));  // compile-time
        float t  = (-c) * pm1;                  // v_mul
        float pn = __builtin_fmaf(x, p, t);     // v_fma
        acc = __builtin_fmaf(wp[n], pn, acc);   // v_fma
        pm1 = p;
        p   = pn;
    }
    // * (1 - x^2)
    return acc * __builtin_fmaf(-x, x, 1.0f);
}

// CDNA5 async global -> LDS copy (VGLOBAL encoding, GV mode), ASYNCcnt-tracked.
// Data is touch-once: non-temporal hint keeps L2 clean for other graph nodes.
__device__ __forceinline__ void async_g2l_b128(unsigned lds_byte_addr, unsigned long long gaddr) {
    asm volatile("global_load_async_to_lds_b128 %0, %1, off th:TH_LOAD_NT"
                 :: "v"(lds_byte_addr), "v"(gaddr)
                 : "memory");
}

__device__ __forceinline__ void wait_async_le2() {
    asm volatile("s_wait_asynccnt 2" ::: "memory");
}
__device__ __forceinline__ void wait_async_0() {
    asm volatile("s_wait_asynccnt 0" ::: "memory");
}

__global__ __launch_bounds__(BLOCK) void legendre1d_kernel(
    const float* __restrict__ x, const float* __restrict__ W,
    float* __restrict__ out, int n) {

    __shared__ float4 lbuf[2][2][BLOCK];   // 16 KB double buffer (320 KB LDS/WGP)

    // Fold monic leading coefficients K_n into the weights (compile-time K_n;
    // lowers to SALU s_mul_f32 on uniform data -> zero VALU cost).
    float wp[NW];
#pragma unroll
    for (int d = 0; d < NW; ++d) {
        double K = 1.0;
#pragma unroll
        for (int m = 2; m <= d; ++m) K *= (2.0 * m - 1.0) / (double)m;  // folded
        wp[d] = W[d] * (float)K;
    }

    const int n8     = n >> 3;             // groups of 8 elements (2 x float4)
    const int tid    = threadIdx.x;
    const int idx    = blockIdx.x * BLOCK + tid;
    const int stride = gridDim.x * BLOCK;

    const float4* __restrict__ x4 = (const float4*)x;
    v4f* __restrict__ o4          = (v4f*)out;

    // Prologue: stage first group.
    if (idx < n8) {
        async_g2l_b128((unsigned)(uintptr_t)&lbuf[0][0][tid],
                       (unsigned long long)(uintptr_t)(x4 + 2 * idx));
        async_g2l_b128((unsigned)(uintptr_t)&lbuf[0][1][tid],
                       (unsigned long long)(uintptr_t)(x4 + 2 * idx + 1));
    }

    int cur = 0;
#pragma unroll 1
    for (int i = idx; i < n8; i += stride) {
        const int inext = i + stride;
        if (inext < n8) {
            async_g2l_b128((unsigned)(uintptr_t)&lbuf[cur ^ 1][0][tid],
                           (unsigned long long)(uintptr_t)(x4 + 2 * inext));
            async_g2l_b128((unsigned)(uintptr_t)&lbuf[cur ^ 1][1][tid],
                           (unsigned long long)(uintptr_t)(x4 + 2 * inext + 1));
            wait_async_le2();              // current group done; next stays in flight
        } else {
            wait_async_0();
        }

        // Each thread reads exactly the slots it staged -> no barrier required.
        float4 xa = lbuf[cur][0][tid];     // ds_load_b128
        float4 xb = lbuf[cur][1][tid];     // ds_load_b128

        v4f ra, rb;                        // 4 independent packed FMA chains
        ra.x = leg_eval(xa.x, wp);
        ra.y = leg_eval(xa.y, wp);
        ra.z = leg_eval(xa.z, wp);
        ra.w = leg_eval(xa.w, wp);
        rb.x = leg_eval(xb.x, wp);
        rb.y = leg_eval(xb.y, wp);
        rb.z = leg_eval(xb.z, wp);
        rb.w = leg_eval(xb.w, wp);

        // Touch-once output: non-temporal b128 stores.
        __builtin_nontemporal_store(ra, o4 + 2 * i);
        __builtin_nontemporal_store(rb, o4 + 2 * i + 1);
        cur ^= 1;
    }

    // Scalar tail (n % 8 elements)
    const int rem = n - (n8 << 3);
    if (blockIdx.x == 0 && tid < rem) {
        const int j = (n8 << 3) + tid;
        out[j] = leg_eval(x[j], wp);
    }
}

extern "C" void kernel_launch(void* const* d_in, const int* in_sizes, int n_in,
                              void* d_out, int out_size, void* d_ws, size_t ws_size,
                              hipStream_t stream) {
    const float* x = (const float*)d_in[0];   // (16, 500000) fp32
    const float* W = (const float*)d_in[1];   // (1, 21) fp32
    float* out     = (float*)d_out;           // (16, 500000) fp32

    const int n  = in_sizes[0];
    const int n8 = n >> 3;
    int tiles = (n8 + BLOCK - 1) / BLOCK;
    if (tiles < 1) tiles = 1;
    const int grid = tiles < 1024 ? tiles : 1024;

    hipLaunchKernelGGL(legendre1d_kernel, dim3(grid), dim3(BLOCK), 0, stream,
                       x, W, out, n);
}